// SoftResidualVQ_31430570672831
// MI455X (gfx1250) — compile-verified
//
#include <hip/hip_runtime.h>
#include <cstdint>
#include <cstddef>

// Problem sizes (fixed by the reference)
#define BQ 32768
#define DQ 512
#define KQ 1024
#define LQ 4

// LDS strides (dwords).
// A tiles: stride 516 (%4==0 for b128 staging, bank = 4*lane + k -> conflict-free b64)
// B chunks are PAIR-INTERLEAVED: Bp[p][2n+b] = B[2p+b][n], so one ds_load_b64
// returns {B[k][n], B[k+1][n]} contiguously. Pair-row stride % 64 == 32 ->
// lane halves (pairs p, p+1) hit disjoint 32-bank halves: conflict-free.
#define LDA   516   // A tile  [32][512]
#define LDSC 1028   // logits/p [32][1024]
#define LDP_S 2080  // scores B chunk rows: 2*1024 padded
#define LDP_Z 1056  // zsoft/whiten B chunk rows: 2*512 padded
#define BUFE  (8 * 1056)  // 8448 floats per buffer >= 4*2080=8320 (double-buffered)

typedef __attribute__((ext_vector_type(2))) float v2f;
typedef __attribute__((ext_vector_type(8))) float v8f;
typedef int v4i __attribute__((vector_size(16)));  // async builtin payload type

#define GLOBAL_AS __attribute__((address_space(1)))
#define LDS_AS __attribute__((address_space(3)))

// ---------------------------------------------------------------------------
// CDNA5 async global->LDS copy (16B), tracked by ASYNCcnt.
// ---------------------------------------------------------------------------
__device__ __forceinline__ void async_b128(const float* g, float* l) {
#if __has_builtin(__builtin_amdgcn_global_load_async_to_lds_b128)
  __builtin_amdgcn_global_load_async_to_lds_b128((GLOBAL_AS v4i*)g,
                                                 (LDS_AS v4i*)l, 0, 0);
#else
  *(float4*)l = *(const float4*)g;
#endif
}

__device__ __forceinline__ void wait_async0() {
#if __has_builtin(__builtin_amdgcn_s_wait_asynccnt)
  __builtin_amdgcn_s_wait_asynccnt(0);
#else
  asm volatile("s_wait_asynccnt 0x0" ::: "memory");
#endif
}

// ---------------------------------------------------------------------------
// WMMA wrapper: D = A(16x4 f32) x B(4x16 f32) + C(16x16 f32)
// ---------------------------------------------------------------------------
__device__ __forceinline__ v8f wmma_f32(v2f a, v2f b, v8f c) {
  return __builtin_amdgcn_wmma_f32_16x16x4_f32(false, a, false, b, (short)0, c,
                                               false, false);
}

// ---------------------------------------------------------------------------
// One K-chunk of the wave-level GEMM. A (row-major) and B (pair-interleaved)
// both live in LDS.
// ---------------------------------------------------------------------------
template <int NT>
__device__ __forceinline__ void gemm_chunk(const float* __restrict__ As, int lda,
                                           int rowbase, int kofs,
                                           const float* __restrict__ Bp, int ldp,
                                           int colbase, int kc, v8f* acc, int lane) {
  const int half = lane >> 4;
  const int lm = lane & 15;
  for (int kk = 0; kk < kc; kk += 4) {
    v2f a = *(const v2f*)(As + (size_t)(rowbase + lm) * lda + kofs + kk + 2 * half);
    const float* bp = Bp + (size_t)((kk >> 1) + half) * ldp + 2 * (colbase + lm);
#pragma unroll
    for (int t = 0; t < NT; ++t) {
      v2f b = *(const v2f*)(bp + 32 * t);
      acc[t] = wmma_f32(a, b, acc[t]);
    }
  }
}

// ---------------------------------------------------------------------------
// prep: pair-interleaved B operands in ws.
//   WP [D/2][2D]:  WP[p][2n+b]  = W[n][2p+b]        (whiten B: B[j][n]=W[n][j])
//   cbP[l][D/2][2K]: cbP[p][2n+b] = cb[l][n][2p+b]  (scores B: B[k][n]=cb[n][k])
//   cbZ[l][K/2][2D]: cbZ[p][2n+b] = cb[l][2p+b][n]  (zsoft  B: B[kk][n]=cb[kk][n])
// ---------------------------------------------------------------------------
__global__ __launch_bounds__(256) void prep_kernel(const float* __restrict__ W,
                                                   const float* __restrict__ cb,
                                                   float* __restrict__ WP,
                                                   float* __restrict__ cbP,
                                                   float* __restrict__ cbZ) {
  size_t i = (size_t)blockIdx.x * 256 + threadIdx.x;
  const size_t wtot = (size_t)DQ * DQ;
  const size_t ptot = (size_t)LQ * DQ * KQ;
  if (i < wtot) {
    size_t p = i / (2 * DQ), c = i % (2 * DQ);
    size_t n = c >> 1, b = c & 1;
    WP[i] = W[n * DQ + 2 * p + b];
  } else if (i < wtot + ptot) {
    size_t t = i - wtot;
    size_t l = t / ((size_t)DQ * KQ), r = t % ((size_t)DQ * KQ);
    size_t p = r / (2 * KQ), c = r % (2 * KQ);
    size_t n = c >> 1, b = c & 1;
    cbP[t] = cb[(l * KQ + n) * DQ + 2 * p + b];
  } else {
    size_t u = i - wtot - ptot;
    if (u < (size_t)LQ * KQ * DQ) {
      size_t l = u / ((size_t)KQ * DQ), r = u % ((size_t)KQ * DQ);
      size_t p = r / (2 * DQ), c = r % (2 * DQ);
      size_t n = c >> 1, b = c & 1;
      cbZ[u] = cb[(l * KQ + 2 * p + b) * DQ + n];
    }
  }
}

// ---------------------------------------------------------------------------
// csq[l*K + k] = sum_d cb[l][k][d]^2   (one wave per row)
// ---------------------------------------------------------------------------
__global__ __launch_bounds__(256) void csq_kernel(const float* __restrict__ cb,
                                                  float* __restrict__ csq) {
  const int lane = threadIdx.x & 31;
  const int wave = threadIdx.x >> 5;
  const int row = blockIdx.x * 8 + wave;
  const float* crow = cb + (size_t)row * DQ;
  float s = 0.f;
  for (int i = lane; i < DQ; i += 32) {
    float v = crow[i];
    s += v * v;
  }
#pragma unroll
  for (int off = 16; off > 0; off >>= 1) s += __shfl_xor(s, off, 32);
  if (lane == 0) csq[row] = s;
}

// ---------------------------------------------------------------------------
// whiten: resid = z0 @ W^T.  Block = 32 rows x 512 cols, double-buffered chunks.
// ---------------------------------------------------------------------------
__global__ __launch_bounds__(256) void whiten_kernel(const float* __restrict__ z0,
                                                     const float* __restrict__ WP,
                                                     float* __restrict__ resid) {
  __shared__ float As[32 * LDA];
  __shared__ float Bs[2][BUFE];
  const int tid = threadIdx.x;
  const int lane = tid & 31;
  const int wave = tid >> 5;
  const size_t rowblk = (size_t)blockIdx.x * 32;

  // Chunk = 8 pair-rows (16 K-dims) of WP, 32 chunks over D=512.
  auto stage = [&](int c, float* buf) {
    const float* src = WP + (size_t)c * 8 * (2 * DQ);
    for (int i = tid; i < 8 * 256; i += 256) {
      const int row = i >> 8, c4 = (i & 255) << 2;
      async_b128(src + (size_t)row * (2 * DQ) + c4, buf + row * LDP_Z + c4);
    }
  };

  // Stage A tile and first B chunk under one wait.
  for (int i = tid; i < 32 * 128; i += 256) {
    const int row = i >> 7, c4 = (i & 127) << 2;
    async_b128(z0 + rowblk * DQ + (size_t)row * DQ + c4, As + row * LDA + c4);
  }
  stage(0, Bs[0]);
  wait_async0();
  __syncthreads();

  const int rowbase = (wave >> 2) * 16;
  const int colbase = (wave & 3) * 128;
  v8f acc[8] = {};
  for (int c = 0; c < 32; ++c) {
    if (c + 1 < 32) stage(c + 1, Bs[(c + 1) & 1]);  // DMA under compute
    gemm_chunk<8>(As, LDA, rowbase, c * 16, Bs[c & 1], LDP_Z, colbase, 16, acc, lane);
    wait_async0();
    __syncthreads();
  }

  const int half = lane >> 4, lm = lane & 15;
#pragma unroll
  for (int t = 0; t < 8; ++t) {
#pragma unroll
    for (int r = 0; r < 8; ++r) {
      const int row = rowbase + r + 8 * half;  // C/D: VGPR r -> M=r / M=8+r
      const int col = colbase + t * 16 + lm;
      resid[(rowblk + row) * DQ + col] = acc[t][r];
    }
  }
}

// ---------------------------------------------------------------------------
// One residual-VQ level, fused per 32-row block:
//   s = resid @ cb^T ; p = softmax((2s - c_sq)/tau) ; z = p @ cb ; resid -= z
// ---------------------------------------------------------------------------
__global__ __launch_bounds__(256) void level_kernel(
    const float* __restrict__ cbP,   // [D/2][2K] this level (scores B)
    const float* __restrict__ cbZ,   // [K/2][2D] this level (zsoft B)
    const float* __restrict__ csq,   // [K] this level
    const float* __restrict__ tau_p, // scalar
    float* __restrict__ resid,       // [B][D] ws, updated in place
    float* __restrict__ zs_out,      // [B][D] slice of d_out
    float* __restrict__ ps_out) {    // [B][K] slice of d_out
  __shared__ float As[32 * LDA];    //  ~64.5 KB staged residual rows
  __shared__ float Sc[32 * LDSC];   // ~128.5 KB logits -> probabilities
  __shared__ float Bs[2][BUFE];     //  ~66 KB double-buffered B chunks
  const int tid = threadIdx.x;
  const int lane = tid & 31;
  const int wave = tid >> 5;
  const int half = lane >> 4, lm = lane & 15;
  const size_t rowblk = (size_t)blockIdx.x * 32;

  // Chunk stagers. Scores: 4 pair-rows (8 K-dims), 64 chunks over D.
  //                Zsoft:  8 pair-rows (16 K-dims), 64 chunks over K.
  auto stage_s = [&](int c, float* buf) {
    const float* src = cbP + (size_t)c * 4 * (2 * KQ);
    for (int i = tid; i < 4 * 512; i += 256) {
      const int row = i >> 9, c4 = (i & 511) << 2;
      async_b128(src + (size_t)row * (2 * KQ) + c4, buf + row * LDP_S + c4);
      __builtin_prefetch(src + (size_t)8 * (2 * KQ) + (size_t)row * (2 * KQ) + c4, 0, 1);
    }
  };
  auto stage_z = [&](int c, float* buf) {
    const float* src = cbZ + (size_t)c * 8 * (2 * DQ);
    for (int i = tid; i < 8 * 256; i += 256) {
      const int row = i >> 8, c4 = (i & 255) << 2;
      async_b128(src + (size_t)row * (2 * DQ) + c4, buf + row * LDP_Z + c4);
    }
  };

  // Stage residual rows and first scores chunk under one wait.
  for (int i = tid; i < 32 * 128; i += 256) {
    const int row = i >> 7, c4 = (i & 127) << 2;
    async_b128(resid + rowblk * DQ + (size_t)row * DQ + c4, As + row * LDA + c4);
  }
  stage_s(0, Bs[0]);
  wait_async0();
  __syncthreads();

  const float inv_tau = 1.0f / tau_p[0];
  const int rowbase = (wave >> 2) * 16;

  // ---- scores GEMM: s[32][1024], double-buffered 8-K chunks ----
  {
    const int colbase = (wave & 3) * 256;  // 16 tiles per wave
    v8f acc[16] = {};
    for (int c = 0; c < 64; ++c) {
      if (c + 1 < 64) stage_s(c + 1, Bs[(c + 1) & 1]);  // DMA under compute
      gemm_chunk<16>(As, LDA, rowbase, c * 8, Bs[c & 1], LDP_S, colbase, 8, acc, lane);
      wait_async0();
      __syncthreads();
    }
    // logits into LDS (z_sq cancels under softmax)
#pragma unroll
    for (int t = 0; t < 16; ++t) {
      const int n = colbase + t * 16 + lm;
      const float cs = csq[n];
#pragma unroll
      for (int r = 0; r < 8; ++r) {
        const int row = rowbase + r + 8 * half;
        Sc[row * LDSC + n] = (2.0f * acc[t][r] - cs) * inv_tau;
      }
    }
  }
  // Issue first zsoft chunk now: its DMA hides under the softmax phase.
  stage_z(0, Bs[0]);
  __syncthreads();

  // ---- softmax over K=1024, 4 rows per wave ----
  for (int rr = 0; rr < 4; ++rr) {
    const int row = wave * 4 + rr;
    float* srow = Sc + row * LDSC;
    float mx = -3.402823466e38f;
#pragma unroll
    for (int i = 0; i < KQ / 32; ++i) mx = fmaxf(mx, srow[lane + 32 * i]);
#pragma unroll
    for (int off = 16; off > 0; off >>= 1) mx = fmaxf(mx, __shfl_xor(mx, off, 32));
    float ss = 0.f;
#pragma unroll
    for (int i = 0; i < KQ / 32; ++i) ss += __expf(srow[lane + 32 * i] - mx);
#pragma unroll
    for (int off = 16; off > 0; off >>= 1) ss += __shfl_xor(ss, off, 32);
    const float inv = 1.0f / ss;
    float* prow = ps_out + (rowblk + row) * KQ;
#pragma unroll
    for (int i = 0; i < KQ / 32; ++i) {
      const int j = lane + 32 * i;
      const float p = __expf(srow[j] - mx) * inv;
      srow[j] = p;   // A operand of the next GEMM
      prow[j] = p;   // output
    }
  }
  wait_async0();
  __syncthreads();

  // ---- z_soft GEMM: z[32][512] = p[32][1024] @ cb[1024][512] ----
  {
    const int colbase = (wave & 3) * 128;  // 8 tiles per wave
    v8f acc[8] = {};
    for (int c = 0; c < 64; ++c) {
      if (c + 1 < 64) stage_z(c + 1, Bs[(c + 1) & 1]);  // DMA under compute
      gemm_chunk<8>(Sc, LDSC, rowbase, c * 16, Bs[c & 1], LDP_Z, colbase, 16, acc, lane);
      wait_async0();
      __syncthreads();
    }
#pragma unroll
    for (int t = 0; t < 8; ++t) {
#pragma unroll
      for (int r = 0; r < 8; ++r) {
        const int row = rowbase + r + 8 * half;
        const int col = colbase + t * 16 + lm;
        const float v = acc[t][r];
        zs_out[(rowblk + row) * DQ + col] = v;
        resid[(rowblk + row) * DQ + col] = As[row * LDA + col] - v;  // old resid
      }
    }
  }
}

// ---------------------------------------------------------------------------
// Host side
// ---------------------------------------------------------------------------
extern "C" void kernel_launch(void* const* d_in, const int* in_sizes, int n_in,
                              void* d_out, int out_size, void* d_ws, size_t ws_size,
                              hipStream_t stream) {
  (void)in_sizes; (void)n_in; (void)out_size; (void)ws_size;
  const float* z0  = (const float*)d_in[0];   // [B, D]
  const float* tau = (const float*)d_in[1];   // [1]
  const float* W   = (const float*)d_in[2];   // [D, D]
  const float* cb  = (const float*)d_in[3];   // [L, K, D]

  float* out = (float*)d_out;
  float* zs = out;                              // [L, B, D]
  float* ps = out + (size_t)LQ * BQ * DQ;       // [L, B, K]

  // Workspace layout (~85 MB):
  float* resid = (float*)d_ws;                              // B*D
  float* cbP   = resid + (size_t)BQ * DQ;                   // L*D*K
  float* cbZ   = cbP + (size_t)LQ * DQ * KQ;                // L*K*D
  float* WP    = cbZ + (size_t)LQ * KQ * DQ;                // D*D
  float* csq   = WP + (size_t)DQ * DQ;                      // L*K

  const size_t prep_total =
      (size_t)DQ * DQ + (size_t)LQ * DQ * KQ + (size_t)LQ * KQ * DQ;
  prep_kernel<<<(unsigned)((prep_total + 255) / 256), 256, 0, stream>>>(W, cb, WP,
                                                                        cbP, cbZ);
  csq_kernel<<<(LQ * KQ) / 8, 256, 0, stream>>>(cb, csq);
  whiten_kernel<<<BQ / 32, 256, 0, stream>>>(z0, WP, resid);
  for (int l = 0; l < LQ; ++l) {
    level_kernel<<<BQ / 32, 256, 0, stream>>>(
        cbP + (size_t)l * DQ * KQ, cbZ + (size_t)l * KQ * DQ, csq + (size_t)l * KQ,
        tau, resid, zs + (size_t)l * BQ * DQ, ps + (size_t)l * BQ * KQ);
  }
}